// PagedAttention_49658411877084
// MI455X (gfx1250) — compile-verified
//
#include <hip/hip_runtime.h>
#include <math.h>

#define BB   32
#define HQN  32
#define HKV  8
#define DD   128
#define GG   4
#define BS   16
#define MAXB 128

#define KSTR 132              // K/Q LDS row stride (floats): conflict-free b64 reads
#define VSTR 144              // V LDS row stride: rows*144 mod 64 banks = {0,16,32,48}
#define PSTR 20               // P LDS row stride
#define PART 520              // floats per partial: 512 out + 4 m + 4 l

#define QS_FL (16 * KSTR)                        // 2112
#define WV_KB (16 * KSTR)                        // one K buffer
#define WV_VB (16 * VSTR)                        // one V buffer
#define WV_PB (16 * PSTR)                        // P buffer
#define WV_FL (2 * WV_KB + 2 * WV_VB + WV_PB)    // 9152 floats per wave
#define SM_FL (QS_FL + 2 * DD + 4 * WV_FL)       // 38976 floats
#define SM_BYTES (SM_FL * 4)                     // 155904 bytes dynamic LDS

typedef float v2f __attribute__((ext_vector_type(2)));
typedef float v8f __attribute__((ext_vector_type(8)));
typedef int   v4i __attribute__((vector_size(4 * sizeof(int))));

#if defined(__AMDGCN__) && __has_builtin(__builtin_amdgcn_global_load_async_to_lds_b128)
#define HAVE_ASYNC_LDS 1
typedef __attribute__((address_space(1))) v4i* glb_v4p;
typedef __attribute__((address_space(3))) v4i* lds_v4p;
#endif

__device__ __forceinline__ v8f wmma_f32(v2f a, v2f b, v8f c) {
  return __builtin_amdgcn_wmma_f32_16x16x4_f32(false, a, false, b, (short)0, c,
                                               false, false);
}

// one wave32 instruction: copy 512B row (16B/lane) global -> LDS, ASYNCcnt-tracked
__device__ __forceinline__ void async_row16(float* l, const float* g) {
#ifdef HAVE_ASYNC_LDS
  __builtin_amdgcn_global_load_async_to_lds_b128((glb_v4p)g, (lds_v4p)l, 0, 0);
#else
  asm volatile("global_load_async_to_lds_b128 %0, %1, off"
               :: "v"((unsigned)(unsigned long long)l),
                  "v"((unsigned long long)g)
               : "memory");
#endif
}

__device__ __forceinline__ void async_wait(void) {
  asm volatile("" ::: "memory");
}

__device__ __forceinline__ void async_wait0() {
#if defined(__AMDGCN__) && __has_builtin(__builtin_amdgcn_s_wait_asynccnt)
  __builtin_amdgcn_s_wait_asynccnt(0);
#else
  asm volatile("s_wait_asynccnt 0x0");
#endif
  asm volatile("" ::: "memory");
}

__device__ __forceinline__ void async_wait16() {
#if defined(__AMDGCN__) && __has_builtin(__builtin_amdgcn_s_wait_asynccnt)
  __builtin_amdgcn_s_wait_asynccnt(16);
#else
  asm volatile("s_wait_asynccnt 0x10");
#endif
  asm volatile("" ::: "memory");
}

__device__ __forceinline__ void async_wait32() {
#if defined(__AMDGCN__) && __has_builtin(__builtin_amdgcn_s_wait_asynccnt)
  __builtin_amdgcn_s_wait_asynccnt(32);
#else
  asm volatile("s_wait_asynccnt 0x20");
#endif
  asm volatile("" ::: "memory");
}

// stage one 16x128 f32 tile (row stride HKV*DD in global) into LDS
__device__ __forceinline__ void stage_tile(float* lbuf, const float* gbase,
                                           int lane, int lstride) {
#pragma unroll
  for (int r = 0; r < 16; ++r)
    async_row16(lbuf + r * lstride + lane * 4,
                gbase + (size_t)r * (HKV * DD) + lane * 4);
  asm volatile("" ::: "memory");   // pin the issue block in program order
}

__launch_bounds__(128)
__global__ void pa_partial(const float* __restrict__ q,
                           const float* __restrict__ nk,
                           const float* __restrict__ nv,
                           const float* __restrict__ kc,
                           const float* __restrict__ vc,
                           const int* __restrict__ bt,
                           const int* __restrict__ cl,
                           float* __restrict__ ws,
                           int nslot) {
  extern __shared__ float smem[];
  float* qs  = smem;               // 16 x KSTR, rows 0..3 = rope(q)*scale, rest 0
  float* nks = smem + QS_FL;       // roped new_k
  float* nvs = nks + DD;           // new_v

  const int bk   = blockIdx.x;
  const int b    = bk >> 3, kh = bk & 7;
  const int tid  = threadIdx.x;
  const int wave = tid >> 5, lane = tid & 31;
  const int ln   = lane & 15, hb = lane >> 4;
  const int pos  = cl[b];
  const int len  = pos + 1;
  const int T    = (len + BS - 1) / BS;
  const int blk_new = bt[b * MAXB + (pos >> 4)];
  const int r_new   = pos & 15;
  const float scale = 0.08838834764831845f;   // 1/sqrt(128)
  const float C0    = 0.14391156831212787f;   // ln(10000)/64

  float* wbase = smem + QS_FL + 2 * DD + wave * WV_FL;
  float* kb0 = wbase;
  float* kb1 = wbase + WV_KB;
  float* vb0 = wbase + 2 * WV_KB;
  float* vb1 = wbase + 2 * WV_KB + WV_VB;
  float* pbf = wbase + 2 * WV_KB + 2 * WV_VB;

  // ---- init LDS ----
  for (int i = 4 * KSTR + tid; i < 16 * KSTR; i += 128) qs[i] = 0.f;  // Q rows 4..15
  for (int i = lane; i < WV_PB; i += 32) pbf[i] = 0.f;                // own P buffer

  for (int e = tid; e < GG * DD; e += 128) {       // RoPE(q) * scale
    int j = e >> 7, d = e & 127;
    int fi = d & 63;
    float ang = (float)pos * __expf(-C0 * (float)fi);
    float sn, cs; sincosf(ang, &sn, &cs);
    const float* qr = q + ((size_t)(b * HQN + kh * GG + j)) * DD;
    float x  = qr[d];
    float xo = (d < 64) ? -qr[d + 64] : qr[d - 64];
    qs[j * KSTR + d] = (x * cs + xo * sn) * scale;
  }
  for (int d = tid; d < DD; d += 128) {            // RoPE(new_k), copy new_v
    int fi = d & 63;
    float ang = (float)pos * __expf(-C0 * (float)fi);
    float sn, cs; sincosf(ang, &sn, &cs);
    const float* kr = nk + ((size_t)(b * HKV + kh)) * DD;
    float x  = kr[d];
    float xo = (d < 64) ? -kr[d + 64] : kr[d - 64];
    nks[d] = x * cs + xo * sn;
    nvs[d] = nv[((size_t)(b * HKV + kh)) * DD + d];
  }
  __syncthreads();

  // ---- hoist Q A-operands into registers (immune to memory clobbers) ----
  v2f qa[32];
  {
    const float* qrow = qs + ln * KSTR + 2 * hb;
#pragma unroll
    for (int cd = 0; cd < 32; ++cd) qa[cd] = *(const v2f*)(qrow + 4 * cd);
  }

  float m_r[4] = {-1e30f, -1e30f, -1e30f, -1e30f};
  float l_r[4] = {0.f, 0.f, 0.f, 0.f};
  v8f acc[8];
#pragma unroll
  for (int dt = 0; dt < 8; ++dt)
    acc[dt] = (v8f){0.f, 0.f, 0.f, 0.f, 0.f, 0.f, 0.f, 0.f};

  const int slot = blockIdx.y * 4 + wave;

  // prologue: async-stage first K and V tiles (K issued first)
  if (slot < T) {
    int blk0 = bt[b * MAXB + slot];
    const float* kbg = kc + (size_t)blk0 * (BS * HKV * DD) + (size_t)kh * DD;
    const float* vbg = vc + (size_t)blk0 * (BS * HKV * DD) + (size_t)kh * DD;
    stage_tile(kb0, kbg, lane, KSTR);
    stage_tile(vb0, vbg, lane, VSTR);
  }

  int it = 0;
  for (int t = slot; t < T; t += nslot, ++it) {
    const int blk = bt[b * MAXB + t];
    float* kbuf = (it & 1) ? kb1 : kb0;
    float* knxt = (it & 1) ? kb0 : kb1;
    float* vbuf = (it & 1) ? vb1 : vb0;
    float* vnxt = (it & 1) ? vb0 : vb1;

    async_wait16();                              // K(t) resident; V(t) may fly
    if (blk == blk_new)                          // virtual insert of roped new_k
      *(float4*)(kbuf + r_new * KSTR + lane * 4) =
          *(const float4*)(nks + lane * 4);

    // issue K(t+1), V(t+1): in-order completion lets us wait partially
    const int  tn = t + nslot;
    const bool has_next = tn < T;
    if (has_next) {
      int blkn = bt[b * MAXB + tn];
      const float* kbg = kc + (size_t)blkn * (BS * HKV * DD) + (size_t)kh * DD;
      const float* vbg = vc + (size_t)blkn * (BS * HKV * DD) + (size_t)kh * DD;
      stage_tile(knxt, kbg, lane, KSTR);
      stage_tile(vnxt, vbg, lane, VSTR);
    }

    // ---- scores = Q * K^T : 32 chained WMMA f32 16x16x4 ----
    v8f c = (v8f){0.f, 0.f, 0.f, 0.f, 0.f, 0.f, 0.f, 0.f};
    const float* krow = kbuf + ln * KSTR + 2 * hb;
#pragma unroll
    for (int cd = 0; cd < 32; ++cd) {
      v2f bbk = *(const v2f*)(krow + 4 * cd);
      c = wmma_f32(qa[cd], bbk, c);
    }

    // ---- flash softmax update (rows 0..3, keys striped over 16 lanes) ----
    const bool valid = (t * BS + ln) < len;
    float alpha[4];
#pragma unroll
    for (int v = 0; v < 4; ++v) {
      float s = valid ? c[v] : -1e30f;
      float x = s;
      x = fmaxf(x, __shfl_xor(x, 1, 32));
      x = fmaxf(x, __shfl_xor(x, 2, 32));
      x = fmaxf(x, __shfl_xor(x, 4, 32));
      x = fmaxf(x, __shfl_xor(x, 8, 32));
      float mn = fmaxf(m_r[v], x);
      alpha[v] = __expf(m_r[v] - mn);
      float pr = __expf(s - mn);
      float rs = pr;
      rs += __shfl_xor(rs, 1, 32);
      rs += __shfl_xor(rs, 2, 32);
      rs += __shfl_xor(rs, 4, 32);
      rs += __shfl_xor(rs, 8, 32);
      l_r[v] = l_r[v] * alpha[v] + rs;
      m_r[v] = mn;
      if (lane < 16) pbf[v * PSTR + ln] = pr;
    }

    // ---- rescale accumulators (rows 4..7 stay exactly zero; skip them) ----
#pragma unroll
    for (int dt = 0; dt < 8; ++dt)
#pragma unroll
      for (int v = 0; v < 4; ++v) acc[dt][v] *= alpha[v];

    // ---- wait until V(t) resident (K/V(t+1) may stay in flight) ----
    if (has_next) async_wait32(); else async_wait0();
    if (blk == blk_new)                          // virtual insert of new_v
      *(float4*)(vbuf + r_new * VSTR + lane * 4) =
          *(const float4*)(nvs + lane * 4);

    // ---- out += P * V : 32 WMMA, all operands from LDS ----
#pragma unroll
    for (int kk = 0; kk < 4; ++kk) {
      v2f a = *(const v2f*)(pbf + ln * PSTR + 4 * kk + 2 * hb);
      const float* v0 = vbuf + (4 * kk + 2 * hb) * VSTR + ln;
#pragma unroll
      for (int dt = 0; dt < 8; ++dt) {
        v2f bbv = { v0[dt * 16], v0[VSTR + dt * 16] };
        acc[dt] = wmma_f32(a, bbv, acc[dt]);
      }
    }
  }

  // ---- write per-wave partial ----
  float* pb = ws + ((size_t)bk * nslot + slot) * PART;
  if (lane < 16) {
#pragma unroll
    for (int dt = 0; dt < 8; ++dt)
#pragma unroll
      for (int v = 0; v < 4; ++v)
        pb[v * DD + dt * 16 + ln] = acc[dt][v];
    if (ln == 0) {
#pragma unroll
      for (int v = 0; v < 4; ++v) { pb[512 + v] = m_r[v]; pb[516 + v] = l_r[v]; }
    }
  }
}

__launch_bounds__(128)
__global__ void pa_combine(const float* __restrict__ ws,
                           float* __restrict__ out, int nslot) {
  const int bk  = blockIdx.x;
  const int tid = threadIdx.x;
  const float* base = ws + (size_t)bk * nslot * PART;
  const int b = bk >> 3, kh = bk & 7;
  for (int e = tid; e < GG * DD; e += 128) {
    int v = e >> 7, d = e & 127;
    float M = -1e30f;
    for (int s = 0; s < nslot; ++s)
      M = fmaxf(M, base[(size_t)s * PART + 512 + v]);
    float num = 0.f, den = 0.f;
    for (int s = 0; s < nslot; ++s) {
      float w = __expf(base[(size_t)s * PART + 512 + v] - M);
      num += w * base[(size_t)s * PART + v * DD + d];
      den += w * base[(size_t)s * PART + 516 + v];
    }
    out[((size_t)(b * HQN + kh * GG + v)) * DD + d] = num / den;
  }
}

extern "C" void kernel_launch(void* const* d_in, const int* in_sizes, int n_in,
                              void* d_out, int out_size, void* d_ws, size_t ws_size,
                              hipStream_t stream) {
  const float* q  = (const float*)d_in[0];
  const float* nk = (const float*)d_in[1];
  const float* nv = (const float*)d_in[2];
  const float* kc = (const float*)d_in[3];
  const float* vc = (const float*)d_in[4];
  const int*   bt = (const int*)d_in[5];
  const int*   cl = (const int*)d_in[6];
  float* ws = (float*)d_ws;

  int NS = 4;  // L-split factor; shrink if workspace is small
  while (NS > 1 &&
         (size_t)BB * HKV * (size_t)(NS * 4) * PART * sizeof(float) > ws_size)
    NS >>= 1;
  const int nslot = NS * 4;

  pa_partial<<<dim3(BB * HKV, NS), 128, SM_BYTES, stream>>>(q, nk, nv, kc, vc,
                                                            bt, cl, ws, nslot);
  pa_combine<<<BB * HKV, 128, 0, stream>>>(ws, (float*)d_out, nslot);
}